// SVFLayer_76072460746907
// MI455X (gfx1250) — compile-verified
//
#include <hip/hip_runtime.h>

// SVF linear time-varying recurrence via chunked associative scan.
//  K1: per-chunk affine transform composition   (32768 lanes)
//  K2: per-batch inter-chunk scan; Kogge-Stone over 4x4 homogeneous
//      matrices implemented with v_wmma_f32_16x16x4_f32 (4 products/WMMA,
//      diagonal blocks), operands staged through LDS per ISA VGPR layouts.
//  K3: per-chunk replay from known start state, emit mixed output.

#define B_BATCH 64
#define N_LEN   32768
#define L_CHUNK 64
#define C_CHUNK (N_LEN / L_CHUNK)   // 512 chunks per batch row
#define GROUP   (C_CHUNK / 32)      // 16 chunks per lane in the scan wave

typedef __attribute__((ext_vector_type(2))) float v2f;
typedef __attribute__((ext_vector_type(8))) float v8f;

// Per-timestep affine step: s_t = S * s_{t-1} + w  (S = 2H - I, w = 2*gHBx)
__device__ __forceinline__ void svf_step_affine(float x, float gg, float tr,
    float& S00, float& S01, float& S10, float& S11, float& w0, float& w1) {
  float T  = 1.0f / fmaf(gg, gg + tr, 1.0f);
  float t2 = 2.0f * T;
  S00 = t2 - 1.0f;
  S01 = -t2 * gg;
  S10 = t2 * gg;
  S11 = fmaf(t2, fmaf(tr, gg, 1.0f), -1.0f);
  w0  = t2 * gg * x;
  w1  = w0 * gg;
}

// ---------------- K1: compose each chunk's affine transform ----------------
__global__ void __launch_bounds__(256) k1_chunk_transform(
    const float* __restrict__ audio, const float* __restrict__ g,
    const float* __restrict__ twoR, float* __restrict__ chunkT) {
  int idx = blockIdx.x * 256 + threadIdx.x;          // one lane per chunk
  int b = idx / C_CHUNK, c = idx % C_CHUNK;
  int base = b * N_LEN + c * L_CHUNK;                // 256B aligned
  const float4* a4 = reinterpret_cast<const float4*>(audio + base);
  const float4* g4 = reinterpret_cast<const float4*>(g + base);
  const float4* r4 = reinterpret_cast<const float4*>(twoR + base);

  float A00 = 1.f, A01 = 0.f, A10 = 0.f, A11 = 1.f, v0 = 0.f, v1 = 0.f;
  for (int j = 0; j < L_CHUNK / 4; ++j) {
    float4 xa = a4[j], xg = g4[j], xr = r4[j];
    float xs[4] = {xa.x, xa.y, xa.z, xa.w};
    float gs[4] = {xg.x, xg.y, xg.z, xg.w};
    float rs[4] = {xr.x, xr.y, xr.z, xr.w};
#pragma unroll
    for (int q = 0; q < 4; ++q) {
      float S00, S01, S10, S11, w0, w1;
      svf_step_affine(xs[q], gs[q], rs[q], S00, S01, S10, S11, w0, w1);
      float nA00 = fmaf(S00, A00, S01 * A10);
      float nA01 = fmaf(S00, A01, S01 * A11);
      float nA10 = fmaf(S10, A00, S11 * A10);
      float nA11 = fmaf(S10, A01, S11 * A11);
      float nv0  = fmaf(S00, v0, fmaf(S01, v1, w0));
      float nv1  = fmaf(S10, v0, fmaf(S11, v1, w1));
      A00 = nA00; A01 = nA01; A10 = nA10; A11 = nA11; v0 = nv0; v1 = nv1;
    }
  }
  float* o = chunkT + idx * 6;
  o[0] = A00; o[1] = A01; o[2] = A10; o[3] = A11; o[4] = v0; o[5] = v1;
}

// ---------------- K2: one wave per batch; WMMA Kogge-Stone scan ------------
__global__ void __launch_bounds__(32) k2_scan_wmma(
    const float* __restrict__ chunkT, float* __restrict__ chunkZ) {
  int b    = blockIdx.x;
  int lane = threadIdx.x;                      // 0..31, full wave
  __shared__ float Q [32][16];                 // 4x4 homogeneous, row-major
  __shared__ float Qn[32][16];
  __shared__ float Dt[16][16];                 // WMMA D staging

  // Phase 1: compose GROUP chunk transforms into this lane's group transform.
  const float* ct = chunkT + (b * C_CHUNK + lane * GROUP) * 6;
  float A00 = 1.f, A01 = 0.f, A10 = 0.f, A11 = 1.f, v0 = 0.f, v1 = 0.f;
  for (int k = 0; k < GROUP; ++k) {
    float S00 = ct[6*k+0], S01 = ct[6*k+1], S10 = ct[6*k+2];
    float S11 = ct[6*k+3], w0  = ct[6*k+4], w1  = ct[6*k+5];
    float nA00 = fmaf(S00, A00, S01 * A10);
    float nA01 = fmaf(S00, A01, S01 * A11);
    float nA10 = fmaf(S10, A00, S11 * A10);
    float nA11 = fmaf(S10, A01, S11 * A11);
    float nv0  = fmaf(S00, v0, fmaf(S01, v1, w0));
    float nv1  = fmaf(S10, v0, fmaf(S11, v1, w1));
    A00 = nA00; A01 = nA01; A10 = nA10; A11 = nA11; v0 = nv0; v1 = nv1;
  }
  // Homogeneous 4x4: [[A v 0],[0 0 1 0],[0 0 0 1]]
  Q[lane][0]  = A00; Q[lane][1]  = A01; Q[lane][2]  = v0;  Q[lane][3]  = 0.f;
  Q[lane][4]  = A10; Q[lane][5]  = A11; Q[lane][6]  = v1;  Q[lane][7]  = 0.f;
  Q[lane][8]  = 0.f; Q[lane][9]  = 0.f; Q[lane][10] = 1.f; Q[lane][11] = 0.f;
  Q[lane][12] = 0.f; Q[lane][13] = 0.f; Q[lane][14] = 0.f; Q[lane][15] = 1.f;
  __syncthreads();

  int m  = lane & 15;        // M/N index within half-wave
  int hh = lane >> 4;        // lane half
  int kb = hh * 2;           // K base for this half (A/B split K across halves)

  // Phase 2: inclusive Kogge-Stone: Q_l <- Q_l * Q_{l-d}
  for (int d = 1; d < 32; d <<= 1) {
    for (int i = 0; i < 8; ++i) {            // 4 matrix products per WMMA
      // A operand (16x4): rows 4p..4p+3 = left matrix Q[4i+p]
      int lA = 4 * i + (m >> 2);
      int rA = m & 3;
      v2f av; av[0] = Q[lA][rA * 4 + kb]; av[1] = Q[lA][rA * 4 + kb + 1];
      // B operand (4x16): cols 4p..4p+3 = right matrix Q[4i+p-d] (or I)
      int p  = m >> 2, cc = m & 3;
      int lB = 4 * i + p - d;
      v2f bv;
      if (lB >= 0) {
        bv[0] = Q[lB][kb * 4 + cc];
        bv[1] = Q[lB][(kb + 1) * 4 + cc];
      } else {
        bv[0] = (kb == cc) ? 1.0f : 0.0f;        // identity
        bv[1] = ((kb + 1) == cc) ? 1.0f : 0.0f;
      }
      v8f acc = {0.f, 0.f, 0.f, 0.f, 0.f, 0.f, 0.f, 0.f};
      v8f dres = __builtin_amdgcn_wmma_f32_16x16x4_f32(
          false, av, false, bv, (short)0, acc, false, false);
      // D layout: VGPR v: lanes 0-15 -> row v, lanes 16-31 -> row v+8
#pragma unroll
      for (int v = 0; v < 8; ++v) Dt[v + 8 * hh][m] = dres[v];
      __syncthreads();
      // Extract the 4 diagonal 4x4 blocks -> Qn[4i+p]
#pragma unroll
      for (int u = 0; u < 2; ++u) {
        int e    = 2 * lane + u;               // 64 elements over 32 lanes
        int pp   = e >> 4;
        int rr   = (e >> 2) & 3;
        int cidx = e & 3;
        Qn[4 * i + pp][rr * 4 + cidx] = Dt[4 * pp + rr][4 * pp + cidx];
      }
      __syncthreads();
    }
#pragma unroll
    for (int t = 0; t < 16; ++t) Q[lane][t] = Qn[lane][t];
    __syncthreads();
  }

  // Phase 3: exclusive prefix -> group start state; replay chunk transforms.
  float P0, P1, P2, P4, P5, P6;
  if (lane == 0) { P0 = 1.f; P1 = 0.f; P2 = 0.f; P4 = 0.f; P5 = 1.f; P6 = 0.f; }
  else {
    P0 = Q[lane-1][0]; P1 = Q[lane-1][1]; P2 = Q[lane-1][2];
    P4 = Q[lane-1][4]; P5 = Q[lane-1][5]; P6 = Q[lane-1][6];
  }
  // s0 = (1,1); homogeneous h = (1,1,1,0)
  float z0 = P0 + P1 + P2;
  float z1 = P4 + P5 + P6;
  float* zo = chunkZ + (b * C_CHUNK + lane * GROUP) * 2;
  for (int k = 0; k < GROUP; ++k) {
    zo[2*k]   = z0;
    zo[2*k+1] = z1;
    float S00 = ct[6*k+0], S01 = ct[6*k+1], S10 = ct[6*k+2];
    float S11 = ct[6*k+3], w0  = ct[6*k+4], w1  = ct[6*k+5];
    float nz0 = fmaf(S00, z0, fmaf(S01, z1, w0));
    float nz1 = fmaf(S10, z0, fmaf(S11, z1, w1));
    z0 = nz0; z1 = nz1;
  }
}

// ---------------- K3: replay each chunk from start state, emit output ------
__global__ void __launch_bounds__(256) k3_emit(
    const float* __restrict__ audio, const float* __restrict__ g,
    const float* __restrict__ twoR, const float* __restrict__ mix,
    const float* __restrict__ chunkZ, float* __restrict__ out) {
  int idx = blockIdx.x * 256 + threadIdx.x;
  int b = idx / C_CHUNK, c = idx % C_CHUNK;
  int base = b * N_LEN + c * L_CHUNK;
  const float4* a4 = reinterpret_cast<const float4*>(audio + base);
  const float4* g4 = reinterpret_cast<const float4*>(g + base);
  const float4* r4 = reinterpret_cast<const float4*>(twoR + base);

  float s0 = chunkZ[idx * 2], s1 = chunkZ[idx * 2 + 1];
  for (int j = 0; j < L_CHUNK / 4; ++j) {
    float4 xa = a4[j], xg = g4[j], xr = r4[j];
    float xs[4] = {xa.x, xa.y, xa.z, xa.w};
    float gs[4] = {xg.x, xg.y, xg.z, xg.w};
    float rs[4] = {xr.x, xr.y, xr.z, xr.w};
#pragma unroll
    for (int q = 0; q < 4; ++q) {
      int   t  = base + j * 4 + q;
      float x  = xs[q], gg = gs[q], tr = rs[q];
      float T  = 1.0f / fmaf(gg, gg + tr, 1.0f);
      float gT = gg * T;
      // Y = gHBx + H * s
      float Y0 = fmaf(T, s0, fmaf(-gT, s1, gT * x));
      float Y1 = fmaf(gT, s0, fmaf(T * fmaf(tr, gg, 1.0f), s1, gT * gg * x));
      s0 = fmaf(2.0f, Y0, -s0);
      s1 = fmaf(2.0f, Y1, -s1);
      float yhp = x - fmaf(tr, Y0, Y1);
      float m0 = mix[3 * t], m1 = mix[3 * t + 1], m2 = mix[3 * t + 2];
      out[t] = fmaf(tr * m0, Y0, fmaf(m1, Y1, m2 * yhp));
    }
  }
}

extern "C" void kernel_launch(void* const* d_in, const int* in_sizes, int n_in,
                              void* d_out, int out_size, void* d_ws, size_t ws_size,
                              hipStream_t stream) {
  const float* audio = (const float*)d_in[0];   // [B,N]
  const float* g     = (const float*)d_in[1];   // [B,N,1]
  const float* twoR  = (const float*)d_in[2];   // [B,N,1]
  const float* mix   = (const float*)d_in[3];   // [B,N,3]
  float* out = (float*)d_out;                   // [B,N]

  float* chunkT = (float*)d_ws;                         // B*C*6 floats
  float* chunkZ = chunkT + B_BATCH * C_CHUNK * 6;       // B*C*2 floats

  int total = B_BATCH * C_CHUNK;                        // 32768 chunks
  k1_chunk_transform<<<total / 256, 256, 0, stream>>>(audio, g, twoR, chunkT);
  k2_scan_wmma<<<B_BATCH, 32, 0, stream>>>(chunkT, chunkZ);
  k3_emit<<<total / 256, 256, 0, stream>>>(audio, g, twoR, mix, chunkZ, out);
}